// ViT_64106681860558
// MI455X (gfx1250) — compile-verified
//
#include <hip/hip_runtime.h>
#include <hip/hip_bf16.h>
#include <math.h>
#include <stdint.h>

typedef float v2f __attribute__((ext_vector_type(2)));
typedef float v8f __attribute__((ext_vector_type(8)));

#define D2048 2048
#define NPATCH 4096
#define NTOK 4097
#define PK 768

// ---------------- CDNA5 async global->LDS copies (ASYNCcnt-tracked) ------
__device__ __forceinline__ void async_to_lds_b128(unsigned lds_off, const float* g) {
    unsigned long long ga = (unsigned long long)(uintptr_t)g;
    asm volatile("global_load_async_to_lds_b128 %0, %1, off"
                 :: "v"(lds_off), "v"(ga) : "memory");
}
__device__ __forceinline__ void async_to_lds_b32(unsigned lds_off, const float* g) {
    unsigned long long ga = (unsigned long long)(uintptr_t)g;
    asm volatile("global_load_async_to_lds_b32 %0, %1, off"
                 :: "v"(lds_off), "v"(ga) : "memory");
}
__device__ __forceinline__ unsigned lds_offset(const void* p) {
    // generic pointer to LDS: addr[31:0] is the LDS offset (aperture rules)
    return (unsigned)(uintptr_t)p;
}

// ---------------------------------------------------------------- patchify
__global__ __launch_bounds__(256) void patchify_k(const float* __restrict__ x,
                                                  float* __restrict__ patches) {
    int idx = blockIdx.x * blockDim.x + threadIdx.x;
    if (idx >= NPATCH * PK) return;
    int p = idx / PK;          // patch id
    int f = idx - p * PK;      // feature id = (i*16 + j)*3 + c
    int py = p >> 6, px = p & 63;
    int c = f % 3;
    int t = f / 3;
    int j = t & 15, i = t >> 4;
    int row = py * 16 + i, col = px * 16 + j;
    patches[idx] = x[((size_t)row * 1024 + col) * 3 + c];
}

// --------------------------------------------------- f32 WMMA tiled GEMM
// C[M,N] = A[M,K] @ B[K,N] + bias ; BM=64 BN=32 BK=16, 256 thr = 8 waves.
// Double-buffered LDS staging via GLOBAL_LOAD_ASYNC_TO_LDS; each wave issues
// 3 async ops per K-tile, so s_wait_asynccnt<=3 pipelines tile i+1 behind i.
__global__ __launch_bounds__(256) void gemm_wmma_f32(
    const float* __restrict__ A, const float* __restrict__ B,
    const float* __restrict__ bias, float* __restrict__ C,
    int M, int N, int K, int lda, int ldb, int ldc) {
    __shared__ float Alds[2][64 * 20];   // [m][k], stride 20 (16B-aligned, conflict-free frags)
    __shared__ float Blds[2][32 * 20];   // [n][k] transposed

    const int t    = threadIdx.x;
    const int lane = t & 31;
    const int wave = t >> 5;       // 0..7
    const int wm   = wave & 3;     // m-tile within block
    const int wn   = wave >> 2;    // n-tile within block
    const int hf   = lane >> 4;    // 0/1 half-wave
    const int lh   = lane & 15;

    const int m0 = blockIdx.y * 64;
    const int n0 = blockIdx.x * 32;

    v8f acc = {};

    // A staging: thread -> (row ar, 4 consecutive k at ac), one async b128
    const int ar = t >> 2;             // 0..63
    const int ac = (t & 3) << 2;       // 0,4,8,12
    // B staging: thread -> col bn, rows bk and bk+8, two async b32 (transposed scatter)
    const int bn = t & 31;
    const int bk = t >> 5;             // 0..7

    // clamp OOB A rows (they only feed C rows that are never stored) -> EXEC stays full
    const int gr = (m0 + ar < M) ? (m0 + ar) : (M - 1);

    const unsigned lds_a0  = lds_offset(&Alds[0][ar * 20 + ac]);
    const unsigned lds_a1  = lds_offset(&Alds[1][ar * 20 + ac]);
    const unsigned lds_b00 = lds_offset(&Blds[0][bn * 20 + bk]);
    const unsigned lds_b01 = lds_offset(&Blds[0][bn * 20 + bk + 8]);
    const unsigned lds_b10 = lds_offset(&Blds[1][bn * 20 + bk]);
    const unsigned lds_b11 = lds_offset(&Blds[1][bn * 20 + bk + 8]);

    const float* gA  = A + (size_t)gr * lda + ac;
    const float* gB0 = B + (size_t)bk * ldb + n0 + bn;
    const float* gB1 = B + (size_t)(bk + 8) * ldb + n0 + bn;

    auto issue = [&](int kb, int sel) {
        async_to_lds_b128(sel ? lds_a1 : lds_a0, gA + kb);
        async_to_lds_b32(sel ? lds_b10 : lds_b00, gB0 + (size_t)kb * ldb);
        async_to_lds_b32(sel ? lds_b11 : lds_b01, gB1 + (size_t)kb * ldb);
    };
    auto consume = [&](int sel) {
        #pragma unroll
        for (int kk = 0; kk < 16; kk += 4) {
            v2f a = *(const v2f*)(&Alds[sel][(wm * 16 + lh) * 20 + kk + 2 * hf]);
            v2f b = *(const v2f*)(&Blds[sel][(wn * 16 + lh) * 20 + kk + 2 * hf]);
            acc = __builtin_amdgcn_wmma_f32_16x16x4_f32(
                false, a, false, b, (short)0, acc, false, false);
        }
    };

    issue(0, 0);
    int sel = 0;
    for (int kb = 0; kb + 16 < K; kb += 16) {
        issue(kb + 16, sel ^ 1);                            // prefetch next tile
        asm volatile("s_wait_asynccnt 0x3" ::: "memory");   // tile `sel` landed (in-order)
        __syncthreads();                                    // ... in every wave
        consume(sel);
        __syncthreads();                                    // frags read; safe to overwrite
        sel ^= 1;
    }
    asm volatile("s_wait_asynccnt 0x0" ::: "memory");
    __syncthreads();
    consume(sel);

    const int col = n0 + wn * 16 + lh;
    const float bv = bias ? bias[col] : 0.0f;
    float* Cw = C + (size_t)(m0 + wm * 16 + 8 * hf) * ldc + col;
    if (m0 + 64 <= M) {                 // full tile: unguarded stores
        #pragma unroll
        for (int r = 0; r < 8; ++r)
            Cw[(size_t)r * ldc] = acc[r] + bv;
    } else {                            // ragged last M-tile
        #pragma unroll
        for (int r = 0; r < 8; ++r) {
            int row = m0 + wm * 16 + r + 8 * hf;
            if (row < M) C[(size_t)row * ldc + col] = acc[r] + bv;
        }
    }
}

// -------------------------------------------------- pos encoding + row 0
__device__ __forceinline__ float pos_enc(int i, int j) {
    int e = ((j & 1) == 0) ? (j / D2048) : ((j + 1) / D2048);
    float denom = powf(10000.0f, (float)e);
    float ang = (float)i / denom;
    return ((j & 1) == 0) ? sinf(ang) : cosf(ang);
}

__global__ __launch_bounds__(256) void finish_vin_k(float* __restrict__ vin,
                                                    const float* __restrict__ cls_tok) {
    int idx = blockIdx.x * blockDim.x + threadIdx.x;
    if (idx >= NTOK * D2048) return;
    int r = idx >> 11;
    int d = idx & (D2048 - 1);
    float p = pos_enc(r, d);
    if (r == 0) vin[idx] = cls_tok[d] + p;
    else        vin[idx] += p;
}

// ----------------------------------------------------------- layer norm
__global__ __launch_bounds__(256) void ln_rows_k(const float* __restrict__ X,
                                                 float* __restrict__ Y,
                                                 const float* __restrict__ g,
                                                 const float* __restrict__ b,
                                                 int ld) {
    __shared__ float s1[256];
    __shared__ float s2[256];
    int row = blockIdx.x, tid = threadIdx.x;
    const float* xr = X + (size_t)row * ld;
    float sum = 0.0f, sq = 0.0f;
    for (int i = tid; i < D2048; i += 256) {
        float v = xr[i];
        sum += v; sq += v * v;
    }
    s1[tid] = sum; s2[tid] = sq;
    __syncthreads();
    for (int s = 128; s > 0; s >>= 1) {
        if (tid < s) { s1[tid] += s1[tid + s]; s2[tid] += s2[tid + s]; }
        __syncthreads();
    }
    float mu   = s1[0] * (1.0f / 2048.0f);
    float var  = s2[0] * (1.0f / 2048.0f) - mu * mu;
    float rstd = rsqrtf(var + 1e-5f);
    float* yr = Y + (size_t)row * ld;
    for (int i = tid; i < D2048; i += 256)
        yr[i] = (xr[i] - mu) * rstd * g[i] + b[i];
}

// -------------------------------------------------- generic vec @ matrix
__device__ __forceinline__ float gelu_exact(float x) {
    return 0.5f * x * (1.0f + erff(x * 0.70710678118654752f));
}

__global__ __launch_bounds__(256) void vecmat_k(const float* __restrict__ vec,
                                                const float* __restrict__ W,
                                                const float* __restrict__ bias,
                                                const float* __restrict__ res,
                                                float* __restrict__ out,
                                                int N, int K, int ldb, int act) {
    int n = blockIdx.x * blockDim.x + threadIdx.x;
    if (n >= N) return;
    float acc = bias[n];
    for (int k = 0; k < K; ++k)
        acc += vec[k] * W[(size_t)k * ldb + n];
    if (act == 1) acc = gelu_exact(acc);
    out[n] = res ? (res[n] + acc) : acc;
}

// ------------------------------------------ attention score row (q0 . K)
__global__ __launch_bounds__(256) void scores_k(const float* __restrict__ q0,
                                                const float* __restrict__ kv,
                                                float* __restrict__ scores,
                                                int nrows) {
    int wave = threadIdx.x >> 5, lane = threadIdx.x & 31;
    int row = blockIdx.x * 8 + wave;
    if (row >= nrows) return;
    const float* kr = kv + (size_t)row * 4096;
    float p = 0.0f;
    for (int k = lane; k < D2048; k += 32) p += q0[k] * kr[k];
    #pragma unroll
    for (int off = 16; off > 0; off >>= 1) p += __shfl_xor(p, off, 32);
    if (lane == 0) scores[row] = p * 0.125f;   // 1/sqrt(64)
}

// ------------------------------------------- single block softmax (+res)
__global__ __launch_bounds__(1024) void softmax_k(const float* __restrict__ in,
                                                  float* __restrict__ out,
                                                  const float* __restrict__ res,
                                                  int n) {
    __shared__ float red[1024];
    int tid = threadIdx.x, bd = blockDim.x;
    float m = -3.402823466e38f;
    for (int i = tid; i < n; i += bd) m = fmaxf(m, in[i]);
    red[tid] = m; __syncthreads();
    for (int s = bd >> 1; s > 0; s >>= 1) {
        if (tid < s) red[tid] = fmaxf(red[tid], red[tid + s]);
        __syncthreads();
    }
    float mx = red[0];
    __syncthreads();
    float sum = 0.0f;
    for (int i = tid; i < n; i += bd) sum += expf(in[i] - mx);
    red[tid] = sum; __syncthreads();
    for (int s = bd >> 1; s > 0; s >>= 1) {
        if (tid < s) red[tid] += red[tid + s];
        __syncthreads();
    }
    float inv = 1.0f / red[0];
    for (int i = tid; i < n; i += bd) {
        float v = expf(in[i] - mx) * inv;
        out[i] = res ? (res[i] + v) : v;
    }
}

// --------------------------------------------------- ctx = attn @ V
__global__ __launch_bounds__(256) void ctx_k(const float* __restrict__ attn,
                                             const float* __restrict__ kv,
                                             float* __restrict__ ctx,
                                             int nrows) {
    int d = blockIdx.x * blockDim.x + threadIdx.x;
    if (d >= D2048) return;
    float acc = 0.0f;
    for (int j = 0; j < nrows; ++j)
        acc += attn[j] * kv[(size_t)j * 4096 + 2048 + d];
    ctx[d] = acc;
}

// ================================================================ launch
extern "C" void kernel_launch(void* const* d_in, const int* in_sizes, int n_in,
                              void* d_out, int out_size, void* d_ws, size_t ws_size,
                              hipStream_t stream) {
    const float* x       = (const float*)d_in[0];
    const float* cls_tok = (const float*)d_in[1];
    const float* W_patch = (const float*)d_in[2];
    const float* b_patch = (const float*)d_in[3];
    const float* ln_g    = (const float*)d_in[4];
    const float* ln_b    = (const float*)d_in[5];
    const float* W_qkv   = (const float*)d_in[6];
    const float* b_qkv   = (const float*)d_in[7];
    const float* W_msa   = (const float*)d_in[8];
    const float* b_msa   = (const float*)d_in[9];
    const float* W_h1    = (const float*)d_in[10];
    const float* b_h1    = (const float*)d_in[11];
    const float* W_h2    = (const float*)d_in[12];
    const float* b_h2    = (const float*)d_in[13];
    const float* W_cls   = (const float*)d_in[14];
    const float* b_cls   = (const float*)d_in[15];
    float* out = (float*)d_out;

    char* ws = (char*)d_ws;
    size_t off = 0;
    auto alloc = [&](size_t bytes) -> float* {
        float* p = (float*)(ws + off);
        off = (off + bytes + 255) & ~(size_t)255;
        return p;
    };
    float* patches = alloc((size_t)NPATCH * PK * 4);
    float* vin     = alloc((size_t)NTOK * D2048 * 4);
    float* ln1     = alloc((size_t)NTOK * D2048 * 4);
    float* kv      = alloc((size_t)NTOK * 4096 * 4);
    float* q0      = alloc(D2048 * 4);
    float* scores  = alloc(NTOK * 4);
    float* ctxv    = alloc(D2048 * 4);
    float* enc1    = alloc(D2048 * 4);
    float* ln2     = alloc(D2048 * 4);
    float* h1      = alloc(1024 * 4);
    float* h2      = alloc(D2048 * 4);
    float* enc2    = alloc(D2048 * 4);
    float* pre     = alloc(1024 * 4);

    // 1) gather patches
    patchify_k<<<(NPATCH * PK + 255) / 256, 256, 0, stream>>>(x, patches);
    // 2) emb = patches @ W_patch + b_patch  -> vin rows 1..4096
    gemm_wmma_f32<<<dim3(D2048 / 32, NPATCH / 64), 256, 0, stream>>>(
        patches, W_patch, b_patch, vin + D2048,
        NPATCH, D2048, PK, PK, D2048, D2048);
    // 3) class token + positional encoding
    finish_vin_k<<<(NTOK * D2048 + 255) / 256, 256, 0, stream>>>(vin, cls_tok);
    // 4) ln1 = LN(vin)
    ln_rows_k<<<NTOK, 256, 0, stream>>>(vin, ln1, ln_g, ln_b, D2048);
    // 5) K,V = ln1 @ W_qkv[:, 2048:6144] + b_qkv[2048:]  (4097 x 4096)
    gemm_wmma_f32<<<dim3(4096 / 32, (NTOK + 63) / 64), 256, 0, stream>>>(
        ln1, W_qkv + 2048, b_qkv + 2048, kv,
        NTOK, 4096, D2048, D2048, 6144, 4096);
    // 6) q0 = ln1[0] @ W_qkv[:, :2048] + b_qkv[:2048]
    vecmat_k<<<(D2048 + 255) / 256, 256, 0, stream>>>(
        ln1, W_qkv, b_qkv, nullptr, q0, D2048, D2048, 6144, 0);
    // 7) scores[j] = q0 . K[j] / 8
    scores_k<<<(NTOK + 7) / 8, 256, 0, stream>>>(q0, kv, scores, NTOK);
    // 8) attn = softmax(scores)  (in-place)
    softmax_k<<<1, 1024, 0, stream>>>(scores, scores, nullptr, NTOK);
    // 9) ctx = attn @ V
    ctx_k<<<(D2048 + 255) / 256, 256, 0, stream>>>(scores, kv, ctxv, NTOK);
    // 10) enc1_0 = vin[0] + ctx @ W_msa + b_msa
    vecmat_k<<<(D2048 + 255) / 256, 256, 0, stream>>>(
        ctxv, W_msa, b_msa, vin, enc1, D2048, D2048, D2048, 0);
    // 11) ln2_0
    ln_rows_k<<<1, 256, 0, stream>>>(enc1, ln2, ln_g, ln_b, D2048);
    // 12) h1 = gelu(ln2_0 @ W_h1 + b_h1)
    vecmat_k<<<(1024 + 255) / 256, 256, 0, stream>>>(
        ln2, W_h1, b_h1, nullptr, h1, 1024, D2048, 1024, 1);
    // 13) h2 = gelu(h1 @ W_h2 + b_h2)
    vecmat_k<<<(D2048 + 255) / 256, 256, 0, stream>>>(
        h1, W_h2, b_h2, nullptr, h2, D2048, 1024, D2048, 1);
    // 14) enc2_0 = enc1_0 + softmax(h2)
    softmax_k<<<1, 1024, 0, stream>>>(h2, enc2, enc1, D2048);
    // 15) pre = enc2_0 @ W_cls + b_cls
    vecmat_k<<<(1000 + 255) / 256, 256, 0, stream>>>(
        enc2, W_cls, b_cls, nullptr, pre, 1000, D2048, 1000, 0);
    // 16) out = softmax(pre)
    softmax_k<<<1, 1024, 0, stream>>>(pre, out, nullptr, 1000);

    (void)in_sizes; (void)n_in; (void)out_size; (void)ws_size;
}